// GCNConv_1580547969919
// MI455X (gfx1250) — compile-verified
//
#include <hip/hip_runtime.h>

#define DFEAT 128

typedef __attribute__((ext_vector_type(2))) float v2f;
typedef __attribute__((ext_vector_type(4))) float v4f;
typedef __attribute__((ext_vector_type(8))) float v8f;

// ---------------------------------------------------------------------------
// Shared GEMM core:  D(16x128 strip) = A(16x128) * W(128x128)  via f32 WMMA.
//
// LDS layout (k-quad interleaved):  Wq[(k>>2)*512 + n*4 + (k&3)] = W[k][n]
// so one ds_load_b128 per (iteration, n-tile) yields both WMMA B operands
// ({W[4q][n],W[4q+1][n]} and {W[4q+2][n],W[4q+3][n]}) in adjacent VGPRs.
//
// K enumeration is permuted so each lane's A fetch is one aligned float4:
//   iteration i: lanes 0-15 load k = 8i+0..3, lanes 16-31 load k = 8i+4..7.
//   WMMA #0 consumes k-set {8i+0,8i+1 | 8i+4,8i+5},
//   WMMA #1 consumes k-set {8i+2,8i+3 | 8i+6,8i+7}.
// B operands use the matching per-half-wave quad, so the reduction is
// identical to the canonical order.
// ---------------------------------------------------------------------------

__device__ __forceinline__ void load_w_quads(const float* __restrict__ W,
                                             float* __restrict__ Wq, int tid)
{
#pragma unroll
    for (int i = 0; i < (DFEAT * DFEAT) / 256; ++i) {
        const int idx = tid + i * 256;          // coalesced global read
        const int k = idx >> 7;
        const int n = idx & 127;
        Wq[((k >> 2) << 9) + (n << 2) + (k & 3)] = W[idx];
    }
}

// ---------------------------------------------------------------------------
// Kernel A: h = x @ W + b
// ---------------------------------------------------------------------------
__global__ __launch_bounds__(256) void gemm_bias_kernel(
    const float* __restrict__ x, const float* __restrict__ W,
    const float* __restrict__ bias, float* __restrict__ h, int N)
{
    __shared__ float Wq[DFEAT * DFEAT];
    const int tid = threadIdx.x;
    load_w_quads(W, Wq, tid);
    __syncthreads();

    const int lane = tid & 31;
    const int wave = tid >> 5;
    const int numTiles = (N + 15) >> 4;
    const int tile = blockIdx.x * 8 + wave;
    if (tile >= numTiles) return;

    const int m0 = tile << 4;
    int arow = m0 + (lane & 15);
    if (arow >= N) arow = N - 1;            // clamped rows are masked on store
    const int hh  = lane >> 4;              // half-wave: 0 or 1
    const int klo = hh << 2;                // 0 or 4
    const int nlo = lane & 15;
    const float* xrow = x + (size_t)arow * DFEAT + klo;

    v8f c[8] = {};

#pragma unroll 2
    for (int i = 0; i < 16; ++i) {
        const float4 a4 = *(const float4*)(xrow + (i << 3));   // 16B aligned
        v2f a01; a01.x = a4.x; a01.y = a4.y;
        v2f a23; a23.x = a4.z; a23.y = a4.w;
        const int q = 2 * i + hh;                 // k-quad index per half-wave
        const float* wrow = Wq + (q << 9) + (nlo << 2);
#pragma unroll
        for (int nt = 0; nt < 8; ++nt) {
            const v4f bq = *(const v4f*)(wrow + (nt << 6));    // ds_load_b128
            v2f b01; b01.x = bq.x; b01.y = bq.y;
            v2f b23; b23.x = bq.z; b23.y = bq.w;
            c[nt] = __builtin_amdgcn_wmma_f32_16x16x4_f32(
                false, a01, false, b01, (short)0, c[nt], false, false);
            c[nt] = __builtin_amdgcn_wmma_f32_16x16x4_f32(
                false, a23, false, b23, (short)0, c[nt], false, false);
        }
    }

    // C/D layout: VGPR r holds (M=r, N=lane) for lanes 0-15, (M=r+8) for 16-31.
    const int rowoff = hh << 3;
    if (m0 + 16 <= N) {                     // fast path: whole tile in bounds
#pragma unroll
        for (int nt = 0; nt < 8; ++nt) {
            const int col = nt * 16 + nlo;
            const float bv = bias[col];
#pragma unroll
            for (int r = 0; r < 8; ++r)
                h[(size_t)(m0 + rowoff + r) * DFEAT + col] = c[nt][r] + bv;
        }
    } else {                                // tail tile only
#pragma unroll
        for (int nt = 0; nt < 8; ++nt) {
            const int col = nt * 16 + nlo;
            const float bv = bias[col];
#pragma unroll
            for (int r = 0; r < 8; ++r) {
                const int row = m0 + rowoff + r;
                if (row < N) h[(size_t)row * DFEAT + col] = c[nt][r] + bv;
            }
        }
    }
}

// ---------------------------------------------------------------------------
// Kernel B: agg[row[e]] += val[e] * h[col[e]]   (one wave per edge)
// h rows are L2-resident (51 MB < 192 MB); atomics resolve at L2.
// ---------------------------------------------------------------------------
__global__ __launch_bounds__(256) void spmm_scatter_kernel(
    const float* __restrict__ h, const float* __restrict__ eval_,
    const int* __restrict__ erow, const int* __restrict__ ecol,
    float* __restrict__ agg, int E)
{
    const int lane = threadIdx.x & 31;
    const long gwave = (long)blockIdx.x * (blockDim.x >> 5) + (threadIdx.x >> 5);
    const long nwave = (long)gridDim.x * (blockDim.x >> 5);
    for (long e = gwave; e < E; e += nwave) {
        const int r = erow[e];
        const int cc = ecol[e];
        const float v = eval_[e];
        const float4 hv = ((const float4*)(h + (size_t)cc * DFEAT))[lane];
        float* op = agg + (size_t)r * DFEAT + lane * 4;
        unsafeAtomicAdd(op + 0, v * hv.x);
        unsafeAtomicAdd(op + 1, v * hv.y);
        unsafeAtomicAdd(op + 2, v * hv.z);
        unsafeAtomicAdd(op + 3, v * hv.w);
    }
}

// ---------------------------------------------------------------------------
// Kernel C: out = relu(agg) + relu(agg) @ Wres + bres   (in place on d_out)
// Row i of the result depends only on row i of agg; each wave reads its 16
// rows fully (K-loop + epilogue) before writing them, so in-place is safe.
// ---------------------------------------------------------------------------
__global__ __launch_bounds__(256) void gemm_res_kernel(
    const float* __restrict__ Wres, const float* __restrict__ bres,
    float* __restrict__ out, int N)
{
    __shared__ float Wq[DFEAT * DFEAT];
    const int tid = threadIdx.x;
    load_w_quads(Wres, Wq, tid);
    __syncthreads();

    const int lane = tid & 31;
    const int wave = tid >> 5;
    const int numTiles = (N + 15) >> 4;
    const int tile = blockIdx.x * 8 + wave;
    if (tile >= numTiles) return;

    const int m0 = tile << 4;
    int arow = m0 + (lane & 15);
    if (arow >= N) arow = N - 1;
    const int hh  = lane >> 4;
    const int klo = hh << 2;
    const int nlo = lane & 15;
    const float* aptr = out + (size_t)arow * DFEAT + klo;

    v8f c[8] = {};

#pragma unroll 2
    for (int i = 0; i < 16; ++i) {
        float4 a4 = *(const float4*)(aptr + (i << 3));
        a4.x = fmaxf(a4.x, 0.0f);           // fused relu on the A operand
        a4.y = fmaxf(a4.y, 0.0f);
        a4.z = fmaxf(a4.z, 0.0f);
        a4.w = fmaxf(a4.w, 0.0f);
        v2f a01; a01.x = a4.x; a01.y = a4.y;
        v2f a23; a23.x = a4.z; a23.y = a4.w;
        const int q = 2 * i + hh;
        const float* wrow = Wq + (q << 9) + (nlo << 2);
#pragma unroll
        for (int nt = 0; nt < 8; ++nt) {
            const v4f bq = *(const v4f*)(wrow + (nt << 6));
            v2f b01; b01.x = bq.x; b01.y = bq.y;
            v2f b23; b23.x = bq.z; b23.y = bq.w;
            c[nt] = __builtin_amdgcn_wmma_f32_16x16x4_f32(
                false, a01, false, b01, (short)0, c[nt], false, false);
            c[nt] = __builtin_amdgcn_wmma_f32_16x16x4_f32(
                false, a23, false, b23, (short)0, c[nt], false, false);
        }
    }

    const int rowoff = hh << 3;
    if (m0 + 16 <= N) {                     // fast path: whole tile in bounds
#pragma unroll
        for (int nt = 0; nt < 8; ++nt) {
            const int col = nt * 16 + nlo;
            const float bv = bres[col];
#pragma unroll
            for (int r = 0; r < 8; ++r) {
                const size_t idx = (size_t)(m0 + rowoff + r) * DFEAT + col;
                const float av = fmaxf(out[idx], 0.0f);   // elementwise +a term
                out[idx] = av + c[nt][r] + bv;
            }
        }
    } else {
#pragma unroll
        for (int nt = 0; nt < 8; ++nt) {
            const int col = nt * 16 + nlo;
            const float bv = bres[col];
#pragma unroll
            for (int r = 0; r < 8; ++r) {
                const int row = m0 + rowoff + r;
                if (row < N) {
                    const size_t idx = (size_t)row * DFEAT + col;
                    const float av = fmaxf(out[idx], 0.0f);
                    out[idx] = av + c[nt][r] + bv;
                }
            }
        }
    }
}

// ---------------------------------------------------------------------------
// Host-side launch. Inputs (setup_inputs order):
// 0:x 1:W 2:b 3:Wres 4:bres 5:edge_val 6:edge_row 7:edge_col. All f32 / i32.
// d_ws holds h (N*128 f32 = 51.2 MB). agg accumulates directly in d_out.
// ---------------------------------------------------------------------------
extern "C" void kernel_launch(void* const* d_in, const int* in_sizes, int n_in,
                              void* d_out, int out_size, void* d_ws, size_t ws_size,
                              hipStream_t stream) {
    const float* x    = (const float*)d_in[0];
    const float* W    = (const float*)d_in[1];
    const float* b    = (const float*)d_in[2];
    const float* Wres = (const float*)d_in[3];
    const float* bres = (const float*)d_in[4];
    const float* ev   = (const float*)d_in[5];
    const int*   erow = (const int*)d_in[6];
    const int*   ecol = (const int*)d_in[7];
    float* out = (float*)d_out;
    float* h   = (float*)d_ws;

    const int N = in_sizes[0] / DFEAT;
    const int E = in_sizes[5];

    // agg accumulates in d_out: zero it (graph-capturable memset node).
    hipMemsetAsync(d_out, 0, (size_t)out_size * sizeof(float), stream);

    const int numTiles = (N + 15) >> 4;
    const int gblocks  = (numTiles + 7) / 8;

    gemm_bias_kernel<<<gblocks, 256, 0, stream>>>(x, W, b, h, N);
    spmm_scatter_kernel<<<2048, 256, 0, stream>>>(h, ev, erow, ecol, out, E);
    gemm_res_kernel<<<gblocks, 256, 0, stream>>>(Wres, bres, out, N);
}